// Linear_27685359190361
// MI455X (gfx1250) — compile-verified
//
#include <hip/hip_runtime.h>

// out = x @ (A@B)^T  ==  (x @ B^T) @ A^T      (rank-16 LoRA factorization)
// x: (16384, 4096) f32   A: (4096, 16) f32   B: (16, 4096) f32
// T = x @ B^T : (16384, 16) f32  (scratch, 1 MB in d_ws)
// out = T @ A^T : (16384, 4096) f32
// HBM traffic: read x (256MB) + write out (256MB)  ->  ~22us at 23.3 TB/s.

typedef float v2f __attribute__((ext_vector_type(2)));
typedef float v8f __attribute__((ext_vector_type(8)));
typedef unsigned int u32x4 __attribute__((ext_vector_type(4)));
typedef int i32x4 __attribute__((ext_vector_type(4)));
typedef int i32x8 __attribute__((ext_vector_type(8)));

#define IN_F   4096
#define OUT_F  4096
#define M_TOT  16384   // 4 * 4096 rows
#define RANK   16

#define CHUNK  128                 // K elements per TDM tile
#define NCHUNK (IN_F / CHUNK)      // 32
#define PITCH  132                 // CHUNK + 4 pad DWORDs (TDM pad) -> bank stride 4
#define BUFSZ  (16 * PITCH)        // floats per LDS tile buffer (2112)
#define BUFB   (BUFSZ * 4)         // bytes per buffer

// ---------------------------------------------------------------------------
// TDM: DMA a 16-row x CHUNK-col f32 tile (row stride 4096 elems) into LDS,
// inserting 4 DWORDs of padding after every 128 DWORDs (-> pitch 132).
// D# fields per CDNA5 ISA.
// ---------------------------------------------------------------------------
__device__ __forceinline__ void tdm_load_tile(const float* gsrc, unsigned lds_byte) {
    unsigned long long ga = (unsigned long long)(size_t)gsrc;
    u32x4 g0;
    g0.x = 1u;                                            // count=1, user D#
    g0.y = lds_byte;                                      // lds_addr
    g0.z = (unsigned)ga;                                  // global_addr[31:0]
    g0.w = (unsigned)((ga >> 32) & 0x01FFFFFFu) | (2u << 30);  // addr[56:32] | type=2
    i32x8 g1;
    g1[0] = (2 << 16)      // data_size = 4B
          | (1 << 20)      // pad_enable
          | (6 << 22)      // pad_interval: 128 DWORDs
          | (3 << 25);     // pad_amount: 4 DWORDs
    g1[1] = (int)(4096u << 16);     // tensor_dim0 = 4096 (low16 in hi half)
    g1[2] = (int)(16384u << 16);    // dim0 hi = 0 | tensor_dim1 = 16384
    g1[3] = (int)((unsigned)CHUNK << 16);  // dim1 hi = 0 | tile_dim0 = CHUNK
    g1[4] = 16;                     // tile_dim1 = 16, tile_dim2 = 0
    g1[5] = 4096;                   // tensor_dim0_stride lo32
    g1[6] = 0;                      // stride0 hi16 | stride1 lo16 (stride1=2^26)
    g1[7] = 0x400;                  // stride1[47:16]
    i32x4 gz = {0, 0, 0, 0};
#if __clang_major__ >= 23
    i32x8 z8 = {0, 0, 0, 0, 0, 0, 0, 0};
    __builtin_amdgcn_tensor_load_to_lds(g0, g1, gz, gz, z8, 0);
#else
    __builtin_amdgcn_tensor_load_to_lds(g0, g1, gz, gz, 0);
#endif
}

// ---------------------------------------------------------------------------
// Stage 1: T[m][r] = sum_k x[m][k] * B[r][k]
// 1024 blocks (one 16-row tile of x), 256 threads = 8 waves.
// Wave 0 TDM-DMAs double-buffered 16xCHUNK tiles of x and B into LDS
// (overlapped via TENSORcnt); all waves run V_WMMA_F32_16X16X4_F32 on
// conflict-free LDS fragments; 8 partial C frags reduced through LDS.
// ---------------------------------------------------------------------------
__global__ __launch_bounds__(256) void lora_stage1(const float* __restrict__ x,
                                                   const float* __restrict__ B,
                                                   float* __restrict__ T) {
    __shared__ float sx[4 * BUFSZ];   // [x ping][x pong][B ping][B pong] 33.8KB

    const int mtile = blockIdx.x;            // 0..1023
    const int tid   = threadIdx.x;
    const int lane  = tid & 31;
    const int w     = tid >> 5;              // wave 0..7
    const int idx16 = lane & 15;             // A-frag M row / B-frag N (=r) col
    const int khalf = (lane < 16) ? 0 : 2;   // K sub-offset per half-wave

    const float* xbase = x + (size_t)mtile * 16 * IN_F;
    const unsigned sxb = (unsigned)(size_t)sx;   // LDS byte offset (addr[31:0])

    if (w == 0) {                            // prologue: chunk 0 (x + B tiles)
        tdm_load_tile(xbase, sxb);
        tdm_load_tile(B,     sxb + 2 * BUFB);
    }

    v8f acc = {};
    for (int c = 0; c < NCHUNK; ++c) {
        if (w == 0) {
            if (c + 1 < NCHUNK) {            // prefetch next pair, wait current
                const unsigned nb = (unsigned)((c + 1) & 1);
                tdm_load_tile(xbase + (c + 1) * CHUNK, sxb + nb * BUFB);
                tdm_load_tile(B     + (c + 1) * CHUNK, sxb + (2 + nb) * BUFB);
                __builtin_amdgcn_s_wait_tensorcnt(2);
            } else {
                __builtin_amdgcn_s_wait_tensorcnt(0);
            }
        }
        __syncthreads();

        const float* bx = sx + (c & 1) * BUFSZ;
        const float* bb = sx + (2 + (c & 1)) * BUFSZ;
        const int kw = w * (CHUNK / 8);      // 16 K-values per wave per chunk
        #pragma unroll
        for (int s = 0; s < CHUNK / 8 / 4; ++s) {   // 4 WMMA steps
            const int k = kw + s * 4 + khalf;
            v2f a = *(const v2f*)(bx + idx16 * PITCH + k);   // ds_load_b64
            v2f b = *(const v2f*)(bb + idx16 * PITCH + k);
            acc = __builtin_amdgcn_wmma_f32_16x16x4_f32(
                      false, a, false, b, (short)0, acc, false, false);
        }
        __syncthreads();
    }

    // Reduce 8 per-wave C fragments via LDS (reuse sx), write 16x16 T tile.
    #pragma unroll
    for (int v = 0; v < 8; ++v) sx[w * 256 + lane * 8 + v] = acc[v];
    __syncthreads();

    float sum = 0.f;
    #pragma unroll
    for (int p = 0; p < 8; ++p) sum += sx[p * 256 + tid];

    // thread t <-> (lane L=t/8, vgpr v=t%8); C layout: M = v + (L<16?0:8), N = L%16
    const int L = tid >> 3;
    const int v = tid & 7;
    T[(size_t)(mtile * 16 + v + ((L < 16) ? 0 : 8)) * RANK + (L & 15)] = sum;
}

// ---------------------------------------------------------------------------
// Stage 2: out[m][i] = sum_r T[m][r] * A[i][r]   (K=16 -> 4 WMMA steps)
// One wave per 16x16 output tile; store-bandwidth bound.
// ---------------------------------------------------------------------------
__global__ __launch_bounds__(256) void lora_stage2(const float* __restrict__ T,
                                                   const float* __restrict__ A,
                                                   float* __restrict__ out) {
    const int tid   = threadIdx.x;
    const int lane  = tid & 31;
    const int w     = tid >> 5;
    const int tile  = blockIdx.x * 8 + w;    // 0..262143
    const int mtile = tile >> 8;
    const int ntile = tile & 255;
    const int idx16 = lane & 15;
    const int khalf = (lane < 16) ? 0 : 2;

    const float* trow = T + (size_t)(mtile * 16 + idx16) * RANK;   // A-matrix = T
    const float* arow = A + (size_t)(ntile * 16 + idx16) * RANK;   // B-matrix = A^T

    v8f c = {};
    #pragma unroll
    for (int s = 0; s < 4; ++s) {
        v2f a = *(const v2f*)(trow + s * 4 + khalf);
        v2f b = *(const v2f*)(arow + s * 4 + khalf);
        c = __builtin_amdgcn_wmma_f32_16x16x4_f32(
                false, a, false, b, (short)0, c, false, false);
    }

    // C/D layout: lane L, vgpr v -> M = v + (L<16?0:8), N = L%16
    float* obase = out + (size_t)(mtile * 16 + ((lane < 16) ? 0 : 8)) * OUT_F
                       + (size_t)(ntile * 16 + idx16);
    #pragma unroll
    for (int v = 0; v < 8; ++v) obase[(size_t)v * OUT_F] = c[v];
}

// ---------------------------------------------------------------------------
extern "C" void kernel_launch(void* const* d_in, const int* in_sizes, int n_in,
                              void* d_out, int out_size, void* d_ws, size_t ws_size,
                              hipStream_t stream) {
    (void)in_sizes; (void)n_in; (void)out_size; (void)ws_size;
    const float* x = (const float*)d_in[0];   // (4, 4096, 4096) -> (16384, 4096)
    const float* A = (const float*)d_in[1];   // (4096, 16)
    const float* B = (const float*)d_in[2];   // (16, 4096)
    float* out = (float*)d_out;               // (16384, 4096)
    float* T   = (float*)d_ws;                // (16384, 16) = 1 MB scratch

    lora_stage1<<<M_TOT / 16, 256, 0, stream>>>(x, B, T);
    lora_stage2<<<(M_TOT / 16) * (OUT_F / 16) / 8, 256, 0, stream>>>(T, A, out);
}